// Birth_Death_Loss_60447369724095
// MI455X (gfx1250) — compile-verified
//
#include <hip/hip_runtime.h>

// Problem constants (all powers of two -> pure shift arithmetic)
#define B_      16
#define C_      8
#define N_      16384            // 2^14
#define BCN     (B_ * C_ * N_)   // 2^21 intervals per set
#define TOTAL   (2 * BCN)        // both interval sets
#define HW_SH   18               // H*W = 512*512 = 2^18
#define W_SH    9                // W = 512 = 2^9

#define BLOCKS  2048
#define THREADS 256

typedef float v2f __attribute__((ext_vector_type(2)));
typedef float v8f __attribute__((ext_vector_type(8)));

__global__ __launch_bounds__(THREADS) void bd_loss_main(
    const float* __restrict__ pred,
    const int*   __restrict__ iv0,   // (B,C,N,2,2) int64 reinterpreted as int pairs
    const int*   __restrict__ iv1,
    float*       __restrict__ partial)
{
    const int  tid    = blockIdx.x * THREADS + threadIdx.x;
    const int  stride = BLOCKS * THREADS;     // 524288 -> exactly 8 intervals/thread

    float sum = 0.0f;

    for (long long i = tid; i < (long long)TOTAL; i += stride) {
        const int       set = (int)(i >> 21);
        const long long j   = i & (long long)(BCN - 1);
        const int       bc  = (int)(j >> 14);           // channel index, [0,128)

        const int4* p = (const int4*)(set ? iv1 : iv0) + (j << 1); // 32B/interval
        // Prefetch next iteration's index data (speculative, OOB-safe)
        __builtin_prefetch((const void*)(p + 2LL * stride), 0, 0);

        int4 a = p[0];   // {h0_lo, h0_hi, w0_lo, w0_hi}
        int4 b = p[1];   // {h1_lo, h1_hi, w1_lo, w1_hi}

        const float* base = pred + ((size_t)bc << HW_SH);
        float birth = base[((unsigned)a.x << W_SH) + (unsigned)a.z];
        float death = base[((unsigned)b.x << W_SH) + (unsigned)b.z];
        float d = birth - death;
        sum += d * d;
    }

    // ---- Wave-level reduction through the matrix unit ----
    // A = ones(16x4), B holds the 32 per-lane partials in its first VGPR.
    // D[m][n] = sum_k B[k][n] (identical for every row m), so d[0] across the
    // wave holds the 16 column sums twice; a 32-lane butterfly of d[0] gives
    // 2 * (wave total), independent of the exact B-operand lane mapping.
    float wsum;
#if __has_builtin(__builtin_amdgcn_wmma_f32_16x16x4_f32)
    {
        v2f amat; amat[0] = 1.0f; amat[1] = 1.0f;
        v2f bmat; bmat[0] = sum;  bmat[1] = 0.0f;
        v8f cacc = {};
        cacc = __builtin_amdgcn_wmma_f32_16x16x4_f32(
            false, amat, false, bmat, (short)0, cacc, false, false);
        float w = cacc[0];
        #pragma unroll
        for (int off = 16; off > 0; off >>= 1)
            w += __shfl_xor(w, off, 32);
        wsum = w * 0.5f;
    }
#else
    {
        float w = sum;
        #pragma unroll
        for (int off = 16; off > 0; off >>= 1)
            w += __shfl_xor(w, off, 32);
        wsum = w;
    }
#endif

    // ---- Block-level reduction (8 waves), fixed order -> deterministic ----
    __shared__ float lds[THREADS / 32];
    const int lane = threadIdx.x & 31;
    const int wave = threadIdx.x >> 5;
    if (lane == 0) lds[wave] = wsum;
    __syncthreads();
    if (threadIdx.x == 0) {
        float t = 0.0f;
        #pragma unroll
        for (int k = 0; k < THREADS / 32; ++k) t += lds[k];
        partial[blockIdx.x] = t;
    }
}

__global__ __launch_bounds__(THREADS) void bd_loss_reduce(
    const float* __restrict__ partial, int n, float* __restrict__ out)
{
    __shared__ float lds[THREADS];
    float s = 0.0f;
    for (int i = threadIdx.x; i < n; i += THREADS) s += partial[i]; // fixed order
    lds[threadIdx.x] = s;
    __syncthreads();
    #pragma unroll
    for (int off = THREADS / 2; off > 0; off >>= 1) {
        if (threadIdx.x < off) lds[threadIdx.x] += lds[threadIdx.x + off];
        __syncthreads();
    }
    if (threadIdx.x == 0) out[0] = lds[0] * (1.0f / (float)B_); // overwrite (no pre-zero needed)
}

extern "C" void kernel_launch(void* const* d_in, const int* in_sizes, int n_in,
                              void* d_out, int out_size, void* d_ws, size_t ws_size,
                              hipStream_t stream) {
    (void)in_sizes; (void)n_in; (void)out_size; (void)ws_size;
    const float* pred = (const float*)d_in[0];
    const int*   iv0  = (const int*)d_in[1];   // int64 data, low dwords carry the index
    const int*   iv1  = (const int*)d_in[2];
    float* partial = (float*)d_ws;             // BLOCKS * 4 bytes = 8 KB
    float* out     = (float*)d_out;

    bd_loss_main<<<BLOCKS, THREADS, 0, stream>>>(pred, iv0, iv1, partial);
    bd_loss_reduce<<<1, THREADS, 0, stream>>>(partial, BLOCKS, out);
}